// FilterLegalMoves_90056874262977
// MI455X (gfx1250) — compile-verified
//
#include <hip/hip_runtime.h>

// FilterLegalMoves: out[i,j] = (j in moves[i] && x[i,j]!=0) ? x[i,j] : -inf
// B=4096 rows, A=10000 cols (f32), K=128 int64 indices per row.
// Strategy: one block per row. Phase 1 fills the row with -inf via b128
// stores while the row's 1KB of indices is DMA'd into LDS with the CDNA5
// async global->LDS path (ASYNCcnt, gfx1250 global_load_async_to_lds_b64).
// Phase 2 gathers only the <=128 legal x values and scatters them over the
// -inf background. Pure data movement: ~164MB store + ~50MB gather read
// ~= 9.5us at 23.3 TB/s (vs 328MB / 14us for a read-everything kernel).

#ifndef A_COLS
#define A_COLS 10000
#endif
#define K_MOVES 128
#define BLOCK 256

__global__ __launch_bounds__(BLOCK) void
FilterLegalMoves_kernel(const float* __restrict__ x,
                        const long long* __restrict__ moves,
                        float* __restrict__ out)
{
    __shared__ long long lds_idx[K_MOVES];

    const int  row     = blockIdx.x;
    const int  tid     = threadIdx.x;
    const long long rowBase = (long long)row * (long long)A_COLS;
    const float NINF   = -__builtin_inff();

    // ---- Kick off async copy of this row's indices (global -> LDS) ----
    if (tid < K_MOVES) {
        const long long moff = (long long)row * K_MOVES + tid;
        // LDS byte address = low 32 bits of the flat pointer to __shared__
        // (flat shared addresses are {aperture_hi32, lds_offset_lo32}).
        unsigned int lds_addr =
            (unsigned int)(unsigned long long)(void*)&lds_idx[tid];
        unsigned long long gaddr = (unsigned long long)(const void*)(moves + moff);
        // GV-mode async 8-byte copy per lane; tracked by ASYNCcnt, overlaps
        // with the -inf fill stores below.
        asm volatile("global_load_async_to_lds_b64 %0, %1, off"
                     :
                     : "v"(lds_addr), "v"(gaddr)
                     : "memory");
    }

    // ---- Phase 1: fill this row of out with -inf using 128-bit stores ----
    // Row byte offset = row * 40000, 16B aligned; A_COLS % 4 == 0.
    float4* out4 = reinterpret_cast<float4*>(out + rowBase);
    const float4 v4 = make_float4(NINF, NINF, NINF, NINF);
    for (int i = tid; i < A_COLS / 4; i += BLOCK) {
        out4[i] = v4;
    }

    // Wait for this wave's async global->LDS transfers to land.
#if __has_builtin(__builtin_amdgcn_s_wait_asynccnt)
    __builtin_amdgcn_s_wait_asynccnt(0);
#else
    asm volatile("s_wait_asynccnt 0x0" ::: "memory");
#endif

    // Order fill stores (L2) before the scatter stores below, workgroup scope.
    __threadfence_block();
    __syncthreads();

    // ---- Phase 2: gather legal x values, scatter over the -inf background ----
    if (tid < K_MOVES) {
        const long long c = lds_idx[tid];          // 0 <= c < A_COLS
        const float v = x[rowBase + c];
        // mask==1 path of the reference: x*1 == 0 (incl. -0.0) -> -inf,
        // NaN propagates (NaN == 0 is false).
        out[rowBase + c] = (v == 0.0f) ? NINF : v;
    }
}

extern "C" void kernel_launch(void* const* d_in, const int* in_sizes, int n_in,
                              void* d_out, int out_size, void* d_ws, size_t ws_size,
                              hipStream_t stream) {
    (void)n_in; (void)out_size; (void)d_ws; (void)ws_size;
    const float*     x     = (const float*)d_in[0];
    const long long* moves = (const long long*)d_in[1];
    float*           out   = (float*)d_out;

    const int B = in_sizes[0] / A_COLS;   // 4096 rows
    FilterLegalMoves_kernel<<<B, BLOCK, 0, stream>>>(x, moves, out);
}